// MPModule_22986664968876
// MI455X (gfx1250) — compile-verified
//
#include <hip/hip_runtime.h>

namespace {

constexpr int Bn = 4;     // batch
constexpr int Dn = 21;    // labels
constexpr int Hn = 512;
constexpr int Wn = 512;

// gfx1250 async copy: global -> LDS, tracked by ASYNCcnt.
__device__ __forceinline__ void async_b32(unsigned lds_addr, const float* g) {
  asm volatile("global_load_async_to_lds_b32 %0, %1, off"
               :: "v"(lds_addr), "v"((unsigned long long)g)
               : "memory");
}
__device__ __forceinline__ void wait_async0() {
  asm volatile("s_wait_asynccnt 0x0" ::: "memory");
}

// wave32 broadcast: lane k's value -> SGPR, usable as scalar VALU operand.
__device__ __forceinline__ float bcast(float v, int k) {
  return __uint_as_float(__builtin_amdgcn_readlane(__float_as_uint(v), k));
}

// One wave (32 lanes) per scan line. Lane l (<21) owns label l and keeps
// V[:,l] in 21 VGPRs; the previous pixel's full label vector lives in 21
// SGPRs, replicated each step with v_readlane (no LDS round-trip).
template<int DIR>
__global__ __launch_bounds__(32)
void sweep(const float* __restrict__ u, const float* __restrict__ ew,
           const float* __restrict__ V, float* __restrict__ out)
{
  __shared__ float wline[Wn];   // async-staged per-line edge weights (T == 512)

  const int lane = threadIdx.x;
  const int lc   = (lane < Dn) ? lane : 0;
  const int line = blockIdx.x;

  constexpr bool horiz = (DIR < 2);          // 0,1: along W ; 2,3: along H
  constexpr bool rev   = (DIR & 1);
  constexpr int  T     = horiz ? Wn : Hn;

  int ubase, wbase, stp;                     // 32-bit offsets (buffers < 2GB)
  if (horiz) {
    const int b = line / Hn, h = line % Hn;
    ubase = ((b * Dn + lc) * Hn + h) * Wn;
    wbase = ((b * 4  + DIR) * Hn + h) * Wn;
    stp   = 1;
  } else {
    const int b = line / Wn, x = line % Wn;
    ubase = ((b * Dn + lc) * Hn) * Wn + x;
    wbase = ((b * 4  + DIR) * Hn) * Wn + x;
    stp   = Wn;
  }

  // Async-stage the whole edge-weight line into LDS (16 copies, 32 lanes each).
  for (int i = lane; i < T; i += 32)
    async_b32((unsigned)(unsigned long long)&wline[i], ew + wbase + i * stp);

  // V column for this lane: Vc[k] = V[k, lane]  (resident in VGPRs)
  float Vc[Dn];
#pragma unroll
  for (int k = 0; k < Dn; ++k) Vc[k] = V[k * Dn + lc];

  wait_async0();

  if (lane < Dn) {                           // EXEC set once for the whole scan
    const int sg  = rev ? -1 : 1;
    const int ds_ = sg * stp;
    const int x0  = rev ? (T - 1) : 0;
    int uidx = ubase + x0 * stp;             // current element offset (per lane)
    int xw   = x0;                           // current scan coordinate

    // Border: L = u
    float L = u[uidx];
    out[uidx] = (DIR == 0) ? fmaf(-3.0f, L, L) : (out[uidx] + L);

    float Lp[Dn];                            // replicated prev label vector (SGPRs)
#pragma unroll
    for (int k = 0; k < Dn; ++k) Lp[k] = bcast(L, k);

    // Prefetch step t = 1.
    float uv_n = u[uidx + ds_];
    float w_n  = wline[xw + sg];
    float o_n  = (DIR == 0) ? 0.0f : out[uidx + ds_];

    for (int t = 1; t < T; ++t) {
      uidx += ds_;
      xw   += sg;
      const float w_c  = w_n;
      const float uv_c = uv_n;
      const float o_c  = o_n;

      // Prefetch t+1 (clamped on the last step; harmless re-read).
      const bool last = (t == T - 1);
      const int  pidx = last ? uidx : uidx + ds_;
      const int  pxw  = last ? xw   : xw + sg;
      uv_n = u[pidx];
      w_n  = wline[pxw];
      if (DIR != 0) o_n = out[pidx];

      // cand[k] = L_prev[k] + w * V[k, lane] : 21 independent FMAs (SGPR addend)
      float c[Dn];
#pragma unroll
      for (int k = 0; k < Dn; ++k) c[k] = fmaf(w_c, Vc[k], Lp[k]);
      // log-depth min tree
#pragma unroll
      for (int off = 1; off < 32; off <<= 1) {
#pragma unroll
        for (int k = 0; k + off < Dn; k += (off << 1))
          c[k] = fminf(c[k], c[k + off]);
      }
      const float Lc = uv_c + c[0];

      out[uidx] = (DIR == 0) ? fmaf(-3.0f, uv_c, Lc) : (o_c + Lc);

      // Replicate new label vector for the next step (readlane, no LDS).
#pragma unroll
      for (int k = 0; k < Dn; ++k) Lp[k] = bcast(Lc, k);
    }
  }
}

} // namespace

extern "C" void kernel_launch(void* const* d_in, const int* in_sizes, int n_in,
                              void* d_out, int out_size, void* d_ws, size_t ws_size,
                              hipStream_t stream) {
  (void)in_sizes; (void)n_in; (void)out_size; (void)d_ws; (void)ws_size;
  const float* u  = (const float*)d_in[0];   // unary         (B,1,D,H,W) f32
  const float* ew = (const float*)d_in[1];   // edge_weights  (B,4,H,W)   f32
  const float* V  = (const float*)d_in[2];   // label_context (D,D)       f32
  float* out = (float*)d_out;                // (B,1,D,H,W) f32

  // Sequential direction kernels: dir0 overwrites, dirs 1-3 accumulate RMW.
  sweep<0><<<Bn * Hn, 32, 0, stream>>>(u, ew, V, out);  // left -> right
  sweep<1><<<Bn * Hn, 32, 0, stream>>>(u, ew, V, out);  // right -> left
  sweep<2><<<Bn * Wn, 32, 0, stream>>>(u, ew, V, out);  // top -> down
  sweep<3><<<Bn * Wn, 32, 0, stream>>>(u, ew, V, out);  // bottom -> up
}